// PartialGNN_62680752718268
// MI455X (gfx1250) — compile-verified
//
#include <hip/hip_runtime.h>

#define LN_EPS 1e-5f

typedef __attribute__((ext_vector_type(2))) float v2f;
typedef __attribute__((ext_vector_type(8))) float v8f;

// ---------------------------------------------------------------------------
// Degree / normalization precompute
// ---------------------------------------------------------------------------
__global__ __launch_bounds__(256) void deg_init_kernel(float* __restrict__ deg, int n) {
  int i = blockIdx.x * blockDim.x + threadIdx.x;
  if (i < n) deg[i] = 1.0f;  // self-loop contributes 1 to every node's degree
}

__global__ __launch_bounds__(256) void deg_accum_kernel(const int* __restrict__ dst,
                                                        float* __restrict__ deg, int E) {
  int e = blockIdx.x * blockDim.x + threadIdx.x;
  if (e < E) atomicAdd(&deg[dst[e]], 1.0f);
}

__global__ __launch_bounds__(256) void dinv_kernel(float* __restrict__ deg, int n) {
  int i = blockIdx.x * blockDim.x + threadIdx.x;
  if (i < n) deg[i] = rsqrtf(fmaxf(deg[i], 1.0f));  // deg -> deg^{-1/2} in place
}

// ---------------------------------------------------------------------------
// hw = h @ W   via V_WMMA_F32_16X16X4_F32
// Block: 256 threads = 8 waves. Wave w computes a 16x16 tile:
//   rows [blockIdx.x*32 + (w>>2)*16, +16), cols [(w&3)*16, +16).
// W (64x64 f32, 16KB) staged in LDS once per block.
// All A/B fragments preloaded into VGPRs, then 16 back-to-back WMMAs.
// Out-of-range rows are clamped (not predicated): their D rows are not stored.
// ---------------------------------------------------------------------------
__global__ __launch_bounds__(256) void gcn_gemm_wmma(const float* __restrict__ h,
                                                     const float* __restrict__ W,
                                                     float* __restrict__ hw, int n) {
  __shared__ float Wlds[64 * 64];
  const int tid = threadIdx.x;

  const float4* W4 = (const float4*)W;
  float4* L4 = (float4*)Wlds;
#pragma unroll
  for (int idx = tid; idx < 1024; idx += 256) L4[idx] = W4[idx];
  __syncthreads();

  const int wave = tid >> 5;
  const int lane = tid & 31;
  const int rowBase = blockIdx.x * 32 + (wave >> 2) * 16;
  const int colBase = (wave & 3) * 16;
  const int hl = lane & 15;   // half-wave lane
  const int hi = lane >> 4;   // 0 = lanes 0-15, 1 = lanes 16-31

  // A fragments: lane (hl,hi) holds rows m = rowBase+hl, K = {k+2*hi, k+2*hi+1}.
  int m = rowBase + hl;
  m = (m < n) ? m : (n - 1);  // clamp: garbage rows feed only unsaved D rows
  const float* arow = h + (size_t)m * 64 + hi * 2;
  v2f a[16];
#pragma unroll
  for (int kk = 0; kk < 16; ++kk) a[kk] = *(const v2f*)(arow + kk * 4);

  // B fragments from LDS: VGPR r of step kk holds K = 4*kk + r + 2*hi.
  const float* bp = Wlds + colBase + hl + hi * 128;  // +(2*hi)*64
  v2f b[16];
#pragma unroll
  for (int kk = 0; kk < 16; ++kk) {
    b[kk][0] = bp[(kk * 4 + 0) * 64];
    b[kk][1] = bp[(kk * 4 + 1) * 64];
  }

  v8f acc = {0.f, 0.f, 0.f, 0.f, 0.f, 0.f, 0.f, 0.f};
#pragma unroll
  for (int kk = 0; kk < 16; ++kk)
    acc = __builtin_amdgcn_wmma_f32_16x16x4_f32(false, a[kk], false, b[kk],
                                                (short)0, acc, false, false);

  // D layout: VGPR r -> row rowBase + r + hi*8, col colBase + hl
  const int col = colBase + hl;
#pragma unroll
  for (int r = 0; r < 8; ++r) {
    const int row = rowBase + r + hi * 8;
    if (row < n) hw[(size_t)row * 64 + col] = acc[r];
  }
}

// ---------------------------------------------------------------------------
// agg[i,:] = hw[i,:] * dinv[i]^2   (self-loop term; also initializes agg)
// ---------------------------------------------------------------------------
__global__ __launch_bounds__(256) void selfloop_kernel(const float* __restrict__ hw,
                                                       const float* __restrict__ dinv,
                                                       float* __restrict__ agg, int n) {
  int idx = blockIdx.x * blockDim.x + threadIdx.x;
  if (idx < n * 64) {
    const float di = dinv[idx >> 6];
    agg[idx] = hw[idx] * (di * di);
  }
}

// ---------------------------------------------------------------------------
// Edge aggregation: one wave32 per edge; each lane moves a float2 of the row.
// Edge metadata is wave-uniform -> force through readfirstlane so the
// compiler uses scalar (SMEM) loads for src/dst/dinv.
// agg[dst,:] += hw[src,:] * dinv[src]*dinv[dst]
// ---------------------------------------------------------------------------
__global__ __launch_bounds__(256) void aggregate_kernel(const float* __restrict__ hw,
                                                        const int* __restrict__ src,
                                                        const int* __restrict__ dst,
                                                        const float* __restrict__ dinv,
                                                        float* __restrict__ agg, int E) {
  const int lane = threadIdx.x & 31;
  int e = __builtin_amdgcn_readfirstlane(blockIdx.x * 8 + (threadIdx.x >> 5));
  if (e >= E) return;
  const int s = __builtin_amdgcn_readfirstlane(src[e]);
  const int d = __builtin_amdgcn_readfirstlane(dst[e]);
  const float w = dinv[s] * dinv[d];
  const v2f v = *(const v2f*)(hw + (size_t)s * 64 + lane * 2);
  float* outp = agg + (size_t)d * 64 + lane * 2;
  atomicAdd(outp + 0, v[0] * w);
  atomicAdd(outp + 1, v[1] * w);
}

// ---------------------------------------------------------------------------
// out[i,:] = relu(layernorm(agg[i,:] + bias) * scale + lnb)
// One wave32 per node; 2 floats per lane; wave shuffles for mean/var.
// ---------------------------------------------------------------------------
__global__ __launch_bounds__(256) void finalize_kernel(const float* __restrict__ agg,
                                                       const float* __restrict__ bias,
                                                       const float* __restrict__ scale,
                                                       const float* __restrict__ lnb,
                                                       float* __restrict__ out, int n) {
  const int wave = threadIdx.x >> 5;
  const int lane = threadIdx.x & 31;
  const int i = blockIdx.x * 8 + wave;
  if (i >= n) return;
  const int j = lane * 2;

  v2f v = *(const v2f*)(agg + (size_t)i * 64 + j);
  const float a0 = v[0] + bias[j];
  const float a1 = v[1] + bias[j + 1];

  float s = a0 + a1;
#pragma unroll
  for (int mm = 16; mm > 0; mm >>= 1) s += __shfl_xor(s, mm, 32);
  const float mu = s * (1.0f / 64.0f);

  const float d0 = a0 - mu;
  const float d1 = a1 - mu;
  float q = d0 * d0 + d1 * d1;
#pragma unroll
  for (int mm = 16; mm > 0; mm >>= 1) q += __shfl_xor(q, mm, 32);
  const float r = rsqrtf(q * (1.0f / 64.0f) + LN_EPS);

  v2f o;
  o[0] = fmaxf(fmaf(d0 * r, scale[j], lnb[j]), 0.0f);
  o[1] = fmaxf(fmaf(d1 * r, scale[j + 1], lnb[j + 1]), 0.0f);
  *(v2f*)(out + (size_t)i * 64 + j) = o;
}

// ---------------------------------------------------------------------------
extern "C" void kernel_launch(void* const* d_in, const int* in_sizes, int n_in,
                              void* d_out, int out_size, void* d_ws, size_t ws_size,
                              hipStream_t stream) {
  const float* x        = (const float*)d_in[0];
  const int*   ei       = (const int*)d_in[1];
  const float* weights  = (const float*)d_in[2];
  const float* biases   = (const float*)d_in[3];
  const float* ln_scale = (const float*)d_in[4];
  const float* ln_bias  = (const float*)d_in[5];

  const int N = in_sizes[0] / 64;
  const int E = in_sizes[1] / 2;
  const int L = in_sizes[2] / (64 * 64);

  const int* src = ei;
  const int* dst = ei + E;

  // Workspace layout (floats): dinv[N] | hw[N*64] | bufA[N*64]
  float* dinv = (float*)d_ws;
  float* hw   = dinv + N;
  float* bufA = hw + (size_t)N * 64;
  float* outF = (float*)d_out;

  const dim3 b256(256);

  deg_init_kernel<<<(N + 255) / 256, b256, 0, stream>>>(dinv, N);
  deg_accum_kernel<<<(E + 255) / 256, b256, 0, stream>>>(dst, dinv, E);
  dinv_kernel<<<(N + 255) / 256, b256, 0, stream>>>(dinv, N);

  const float* h_in = x;
  for (int l = 0; l < L; ++l) {
    float* h_out = (l == L - 1) ? outF : bufA;  // aggregate + finalize in place

    gcn_gemm_wmma<<<(N + 31) / 32, b256, 0, stream>>>(
        h_in, weights + (size_t)l * 64 * 64, hw, N);

    selfloop_kernel<<<(N * 64 + 255) / 256, b256, 0, stream>>>(hw, dinv, h_out, N);

    aggregate_kernel<<<(E + 7) / 8, b256, 0, stream>>>(
        hw, src, dst, dinv, h_out, E);

    finalize_kernel<<<(N + 7) / 8, b256, 0, stream>>>(
        h_out, biases + (size_t)l * 64, ln_scale + (size_t)l * 64,
        ln_bias + (size_t)l * 64, h_out, N);

    h_in = h_out;
  }
}